// AngleFreqEnhance_29686813950003
// MI455X (gfx1250) — compile-verified
//
#include <hip/hip_runtime.h>
#include <math.h>

typedef float v2f __attribute__((ext_vector_type(2)));
typedef float v8f __attribute__((ext_vector_type(8)));

constexpr int   NN    = 256;
constexpr int   NPIX  = NN * NN;
constexpr int   XP    = 257;                 // padded LDS pitch (breaks 64-bank stride-256 conflicts)
constexpr float PI_F  = 3.14159265358979323846f;

__device__ __forceinline__ v8f wmma4(v2f a, v2f b, v8f c) {
  // D(16x16,f32) = A(16x4,f32) * B(4x16,f32) + C
  return __builtin_amdgcn_wmma_f32_16x16x4_f32(false, a, false, b, (short)0, c,
                                               false, false);
}
__device__ __forceinline__ v8f vz8() {
  v8f z = {0.f, 0.f, 0.f, 0.f, 0.f, 0.f, 0.f, 0.f};
  return z;
}
// K-pair-packed constant fragment load: one aligned global_load_b64 yielding
// {F[2*kh][col], F[2*kh+1][col]} — exactly one WMMA operand register pair.
__device__ __forceinline__ v2f ldp(const float* __restrict__ p, int kh, int col) {
  return *(const v2f*)(p + (((size_t)kh * NN + col) << 1));
}

// ---------------------------------------------------------------------------
// K0: build ortho DFT matrix F = A + i*Bm (A=cos/16, Bm=-sin/16) in K-pair-
// packed layout, plus the ifftshift'ed angular gain G'.
// F is symmetric, so the same packed buffer serves A-side and B-side frags.
// ---------------------------------------------------------------------------
__global__ void init_consts(const float* __restrict__ w, float* __restrict__ APk,
                            float* __restrict__ BPk, float* __restrict__ Gc) {
  const int i = blockIdx.x;    // 0..255  (K index in packed layout)
  const int k = threadIdx.x;   // 0..255  (column)
  // twiddle: exp(-2*pi*i*k/256)/16, argument reduced mod 256 for accuracy
  const int t = (i * k) & 255;
  float s, c;
  sincosf((float)t * (PI_F / 128.0f), &s, &c);
  const size_t pidx = ((size_t)(i >> 1) * NN + k) * 2 + (i & 1);
  APk[pidx] = c * (1.0f / 16.0f);
  BPk[pidx] = -s * (1.0f / 16.0f);
  // gain at UNshifted freq (i,k): shifted coords relative to center 128
  const float y  = (float)(((i + 128) & 255) - 128);
  const float x  = (float)(((k + 128) & 255) - 128);
  const float r  = sqrtf(y * y + x * x);
  const float th = atan2f(y, x) + PI_F;                  // [0, 2*pi]
  int ai = (int)floorf(th * (8.0f / (2.0f * PI_F)));
  ai &= 7;
  Gc[i * NN + k] = (r > 0.3f * 128.0f) ? w[ai] : 1.0f;
}

// ---------------------------------------------------------------------------
// K1: per-image fused forward:  V = conj(F) * (G' ⊙ (F * X * F))
// X cached in LDS (256x257).  For each 16-wide column panel:
//   B: T = X * F[:,panel]        (real * complex)
//   C: Q = G' ⊙ (F * T)          (complex * complex)  -> written over T in LDS
//   D: V = conj(F) * Q           -> panel written to global workspace
// 8 waves, each owns two 16-row tiles; K accumulated 4 at a time via WMMA.
// ---------------------------------------------------------------------------
__global__ void __launch_bounds__(256) angle_fft_fwd(
    const float* __restrict__ X, const float* __restrict__ APk,
    const float* __restrict__ BPk, const float* __restrict__ Gc,
    float* __restrict__ U) {
  extern __shared__ float smem[];
  float* Xs = smem;                 // NN * XP  (257 KB)
  float* Tr = smem + NN * XP;       // NN * 16  (16 KB)
  float* Ti = Tr + NN * 16;         // NN * 16  (16 KB)

  const int img  = blockIdx.x;
  const int tid  = threadIdx.x;
  const int lane = tid & 31;
  const int wv   = tid >> 5;        // 0..7
  const int fm   = lane & 15;       // fragment row/col within tile
  const int hi   = lane >> 4;       // lane half (K/M striping)
  const int m0   = wv * 32;         // this wave's two row tiles: m0, m0+16

  const float* Ximg = X + (size_t)img * NPIX;
  float* Urp = U + (size_t)img * (2 * NPIX);
  float* Uip = Urp + NPIX;

  for (int idx = tid; idx < NPIX; idx += 256)
    Xs[(idx >> 8) * XP + (idx & 255)] = Ximg[idx];
  __syncthreads();

  for (int jp = 0; jp < 16; ++jp) {
    const int j0 = jp * 16;

    // ---- Phase B: T = Xs * F[:, j0..j0+15]
    v8f tR0 = vz8(), tI0 = vz8(), tR1 = vz8(), tI1 = vz8();
#pragma unroll 2
    for (int k0 = 0; k0 < NN; k0 += 4) {
      const int ka = k0 + 2 * hi;
      const int kh = ka >> 1;
      v2f a0, a1;
      a0.x = Xs[(m0 + fm) * XP + ka];       a0.y = Xs[(m0 + fm) * XP + ka + 1];
      a1.x = Xs[(m0 + 16 + fm) * XP + ka];  a1.y = Xs[(m0 + 16 + fm) * XP + ka + 1];
      const v2f br = ldp(APk, kh, j0 + fm);
      const v2f bi = ldp(BPk, kh, j0 + fm);
      tR0 = wmma4(a0, br, tR0);  tI0 = wmma4(a0, bi, tI0);
      tR1 = wmma4(a1, br, tR1);  tI1 = wmma4(a1, bi, tI1);
    }
    __syncthreads();                       // prior phase-D reads of T are done
    for (int v = 0; v < 8; ++v) {
      const int r = m0 + v + 8 * hi;
      Tr[r * 16 + fm]        = tR0[v];  Ti[r * 16 + fm]        = tI0[v];
      Tr[(r + 16) * 16 + fm] = tR1[v];  Ti[(r + 16) * 16 + fm] = tI1[v];
    }
    __syncthreads();

    // ---- Phase C: Q = G' ⊙ (F * T)   (Qr = A*Tr - Bm*Ti, Qi = A*Ti + Bm*Tr)
    v8f qR0 = vz8(), qI0 = vz8(), qR1 = vz8(), qI1 = vz8();
#pragma unroll 2
    for (int k0 = 0; k0 < NN; k0 += 4) {
      const int ka = k0 + 2 * hi;
      const int kh = ka >> 1;
      // F symmetric: F[m][k] == F[k][m] -> packed coalesced A-fragment loads
      const v2f fr0 = ldp(APk, kh, m0 + fm);
      const v2f fi0 = ldp(BPk, kh, m0 + fm);
      const v2f fr1 = ldp(APk, kh, m0 + 16 + fm);
      const v2f fi1 = ldp(BPk, kh, m0 + 16 + fm);
      v2f tr, ti;
      tr.x = Tr[ka * 16 + fm];  tr.y = Tr[(ka + 1) * 16 + fm];
      ti.x = Ti[ka * 16 + fm];  ti.y = Ti[(ka + 1) * 16 + fm];
      const v2f fn0 = -fi0, fn1 = -fi1;    // f32 WMMA has no A/B NEG -> negate operand
      qR0 = wmma4(fr0, tr, qR0);  qR0 = wmma4(fn0, ti, qR0);
      qI0 = wmma4(fr0, ti, qI0);  qI0 = wmma4(fi0, tr, qI0);
      qR1 = wmma4(fr1, tr, qR1);  qR1 = wmma4(fn1, ti, qR1);
      qI1 = wmma4(fr1, ti, qI1);  qI1 = wmma4(fi1, tr, qI1);
    }
    for (int v = 0; v < 8; ++v) {          // apply angular gain
      const int u = m0 + v + 8 * hi;
      const float g0 = Gc[u * NN + j0 + fm];
      const float g1 = Gc[(u + 16) * NN + j0 + fm];
      qR0[v] *= g0;  qI0[v] *= g0;
      qR1[v] *= g1;  qI1[v] *= g1;
    }
    __syncthreads();                       // phase-C reads of T are done
    for (int v = 0; v < 8; ++v) {          // overwrite T with Q (no extra LDS)
      const int r = m0 + v + 8 * hi;
      Tr[r * 16 + fm]        = qR0[v];  Ti[r * 16 + fm]        = qI0[v];
      Tr[(r + 16) * 16 + fm] = qR1[v];  Ti[(r + 16) * 16 + fm] = qI1[v];
    }
    __syncthreads();

    // ---- Phase D: V = conj(F) * Q  (Vr = A*Qr + Bm*Qi, Vi = A*Qi - Bm*Qr)
    v8f uR0 = vz8(), uI0 = vz8(), uR1 = vz8(), uI1 = vz8();
#pragma unroll 2
    for (int k0 = 0; k0 < NN; k0 += 4) {
      const int ka = k0 + 2 * hi;
      const int kh = ka >> 1;
      const v2f fr0 = ldp(APk, kh, m0 + fm);
      const v2f fi0 = ldp(BPk, kh, m0 + fm);
      const v2f fr1 = ldp(APk, kh, m0 + 16 + fm);
      const v2f fi1 = ldp(BPk, kh, m0 + 16 + fm);
      v2f tr, ti;
      tr.x = Tr[ka * 16 + fm];  tr.y = Tr[(ka + 1) * 16 + fm];
      ti.x = Ti[ka * 16 + fm];  ti.y = Ti[(ka + 1) * 16 + fm];
      const v2f fn0 = -fi0, fn1 = -fi1;
      uR0 = wmma4(fr0, tr, uR0);  uR0 = wmma4(fi0, ti, uR0);
      uI0 = wmma4(fr0, ti, uI0);  uI0 = wmma4(fn0, tr, uI0);
      uR1 = wmma4(fr1, tr, uR1);  uR1 = wmma4(fi1, ti, uR1);
      uI1 = wmma4(fr1, ti, uI1);  uI1 = wmma4(fn1, tr, uI1);
    }
    for (int v = 0; v < 8; ++v) {
      const int r = m0 + v + 8 * hi;
      Urp[r * NN + j0 + fm]        = uR0[v];  Uip[r * NN + j0 + fm]        = uI0[v];
      Urp[(r + 16) * NN + j0 + fm] = uR1[v];  Uip[(r + 16) * NN + j0 + fm] = uI1[v];
    }
    __syncthreads();
  }
}

// ---------------------------------------------------------------------------
// K2: out = X + Re(V * conj(F)) = X + Vr*A + Vi*Bm, by 16-row panels.
// ---------------------------------------------------------------------------
__global__ void __launch_bounds__(256) angle_fft_inv(
    const float* __restrict__ X, const float* __restrict__ APk,
    const float* __restrict__ BPk, const float* __restrict__ U,
    float* __restrict__ out) {
  __shared__ float Vr[16 * XP];
  __shared__ float Vi[16 * XP];
  const int img  = blockIdx.x;
  const int tid  = threadIdx.x;
  const int lane = tid & 31;
  const int wv   = tid >> 5;
  const int fm   = lane & 15;
  const int hi   = lane >> 4;
  const int n0   = wv * 32;

  const float* Urp  = U + (size_t)img * (2 * NPIX);
  const float* Uip  = Urp + NPIX;
  const float* Ximg = X + (size_t)img * NPIX;
  float*       Oimg = out + (size_t)img * NPIX;

  for (int ip = 0; ip < 16; ++ip) {
    const int r0 = ip * 16;
    for (int idx = tid; idx < 16 * NN; idx += 256) {
      const int r = idx >> 8, c = idx & 255;
      Vr[r * XP + c] = Urp[(r0 + r) * NN + c];
      Vi[r * XP + c] = Uip[(r0 + r) * NN + c];
    }
    __syncthreads();
    v8f o0 = vz8(), o1 = vz8();
#pragma unroll 2
    for (int k0 = 0; k0 < NN; k0 += 4) {
      const int ka = k0 + 2 * hi;
      const int kh = ka >> 1;
      v2f ar, ac;
      ar.x = Vr[fm * XP + ka];  ar.y = Vr[fm * XP + ka + 1];
      ac.x = Vi[fm * XP + ka];  ac.y = Vi[fm * XP + ka + 1];
      const v2f b0r = ldp(APk, kh, n0 + fm);
      const v2f b0i = ldp(BPk, kh, n0 + fm);
      const v2f b1r = ldp(APk, kh, n0 + 16 + fm);
      const v2f b1i = ldp(BPk, kh, n0 + 16 + fm);
      o0 = wmma4(ar, b0r, o0);  o0 = wmma4(ac, b0i, o0);
      o1 = wmma4(ar, b1r, o1);  o1 = wmma4(ac, b1i, o1);
    }
    for (int v = 0; v < 8; ++v) {
      const int row = r0 + v + 8 * hi;
      const int c0 = n0 + fm, c1 = n0 + 16 + fm;
      Oimg[row * NN + c0] = Ximg[row * NN + c0] + o0[v];
      Oimg[row * NN + c1] = Ximg[row * NN + c1] + o1[v];
    }
    __syncthreads();
  }
}

// ---------------------------------------------------------------------------
extern "C" void kernel_launch(void* const* d_in, const int* in_sizes, int n_in,
                              void* d_out, int out_size, void* d_ws, size_t ws_size,
                              hipStream_t stream) {
  const float* x = (const float*)d_in[0];       // [4,256,256,256] f32
  const float* w = (const float*)d_in[1];       // [8] f32 angle weights
  const int nimg = in_sizes[0] / NPIX;          // 1024 images

  // workspace: complex intermediate V (nimg*2*NPIX f32 ~= 512 MB), then consts
  float* U   = (float*)d_ws;
  float* APk = U + (size_t)nimg * 2 * NPIX;     // K-pair-packed cos/16
  float* BPk = APk + NPIX;                      // K-pair-packed -sin/16
  float* Gc  = BPk + NPIX;                      // ifftshifted gain

  init_consts<<<NN, NN, 0, stream>>>(w, APk, BPk, Gc);

  const size_t lds_fwd = (size_t)(NN * XP + 2 * NN * 16) * sizeof(float); // ~289 KB of 320 KB WGP LDS
  hipFuncSetAttribute((const void*)angle_fft_fwd,
                      hipFuncAttributeMaxDynamicSharedMemorySize, (int)lds_fwd);

  angle_fft_fwd<<<nimg, 256, lds_fwd, stream>>>(x, APk, BPk, Gc, U);
  angle_fft_inv<<<nimg, 256, 0, stream>>>(x, APk, BPk, U, (float*)d_out);
}